// TeacherSignalSpeaker_88304527606622
// MI455X (gfx1250) — compile-verified
//
#include <hip/hip_runtime.h>
#include <stdint.h>

// ---------------------------------------------------------------------------
// TeacherSignalSpeaker: B x 24 f32 -> (comm B x 48, state B x 32), HBM-bound
// (416 B/row, ~18 us floor at 23.3 TB/s). Input staged via CDNA5 async
// global->LDS loads (ASYNCcnt path, non-temporal); compute uses gfx1250
// hardware transcendentals (v_tanh_f32, v_sqrt_f32, v_rcp_f32); outputs
// streamed with non-temporal 128-bit stores.
// ---------------------------------------------------------------------------

typedef float v4f __attribute__((ext_vector_type(4)));

#define BLK     256
#define ROW_F   24          // floats per input row
#define SLOT_V4 7           // 7 float4 slots per thread in LDS (112 B, padded)

// gfx1250 hardware tanh (TRANS32 op). Two v_nops inside the blob cover the
// documented TRANS co-execution hazard before the result may be consumed.
__device__ __forceinline__ float htanh(float x) {
    float y;
    asm("v_tanh_f32 %0, %1\n\t"
        "v_nop\n\t"
        "v_nop"
        : "=v"(y) : "v"(x));
    return y;
}

__device__ __forceinline__ float fsqrt(float x) { return __builtin_amdgcn_sqrtf(x); }
__device__ __forceinline__ float frcp(float x)  { return __builtin_amdgcn_rcpf(x); }

__global__ __launch_bounds__(BLK) void speaker_kernel(const float* __restrict__ in,
                                                      float* __restrict__ comm,
                                                      float* __restrict__ state,
                                                      int B)
{
    __shared__ v4f tile[BLK * SLOT_V4];     // 28 KB per block

    const int t = threadIdx.x;
    const int r = blockIdx.x * BLK + t;
    if (r >= B) return;

    // ---- async global -> LDS staging of this lane's 96-byte row -----------
    const float* rowp = in + (size_t)r * ROW_F;
    // addrspace(3) offset: low 32 bits of the flat shared-aperture address
    unsigned lbase = (unsigned)(unsigned long long)(uintptr_t)&tile[t * SLOT_V4];
#pragma unroll
    for (int k = 0; k < 6; ++k) {
        unsigned ldst = lbase + 16u * (unsigned)k;
        const float* gsrc = rowp + 4 * k;
        asm volatile("global_load_async_to_lds_b128 %0, %1, off th:TH_LOAD_NT"
                     :
                     : "v"(ldst), "v"(gsrc)
                     : "memory");
    }
    asm volatile("s_wait_asynccnt 0" ::: "memory");

    // ---- pull the 24 floats back into registers ---------------------------
    float s[24];
#pragma unroll
    for (int k = 0; k < 6; ++k) {
        v4f v = tile[t * SLOT_V4 + k];
        s[4 * k + 0] = v.x;
        s[4 * k + 1] = v.y;
        s[4 * k + 2] = v.z;
        s[4 * k + 3] = v.w;
    }

    float px[4], py[4], vx[4], vy[4], q[4];
#pragma unroll
    for (int i = 0; i < 4; ++i) {
        px[i] = s[5 * i + 0];
        py[i] = s[5 * i + 1];
        vx[i] = s[5 * i + 2];
        vy[i] = s[5 * i + 3];
        q[i]  = s[5 * i + 4];
    }
    const float gx = s[20], gy = s[21], gq = s[22], gm = s[23];

    // goal direction (normalized with eps clamp); rcp instead of divide
    const float inv = frcp(fmaxf(fsqrt(gx * gx + gy * gy), 1e-12f));
    const float gpx = gx * inv, gpy = gy * inv;
    const float sgn = (gm > 0.0f) ? 1.0f : ((gm < 0.0f) ? -1.0f : 0.0f);

    // anchor = mean position
    const float ax = 0.25f * (px[0] + px[1] + px[2] + px[3]);
    const float ay = 0.25f * (py[0] + py[1] + py[2] + py[3]);

    // pairwise distances: symmetric, diag masked to exactly 100.0
    // (dist(i,i)=0, +EYE_BIG) -> only 6 sqrts needed
    float dist[4][4];
#pragma unroll
    for (int i = 0; i < 4; ++i) dist[i][i] = 100.0f;
#pragma unroll
    for (int i = 0; i < 4; ++i)
#pragma unroll
        for (int j = i + 1; j < 4; ++j) {
            float dx = px[i] - px[j];
            float dy = py[i] - py[j];
            float d  = fsqrt(dx * dx + dy * dy);
            dist[i][j] = d;
            dist[j][i] = d;
        }

    float score[4];
#pragma unroll
    for (int i = 0; i < 4; ++i) {
        // argmin over masked distances, first-min tie rule (strict <),
        // carrying nearest velocity via selects (no dynamic reg indexing)
        float best = dist[i][0];
        float nvx = vx[0], nvy = vy[0];
#pragma unroll
        for (int j = 1; j < 4; ++j) {
            float dj = dist[i][j];
            bool c = dj < best;
            best = c ? dj : best;
            nvx  = c ? vx[j] : nvx;
            nvy  = c ? vy[j] : nvy;
        }
        float va  = (vx[i] * nvx + vy[i] * nvy) * gm;
        float rx  = px[i] - ax, ry = py[i] - ay;
        float rel = -fsqrt(rx * rx + ry * ry) * sgn;
        float ps  = px[i] * gpx + py[i] * gpy;
        score[i]  = 0.25f * ps + 1.15f * (q[i] * gq) + 0.9f * va + 0.15f * rel;
    }

    // standardize over the 4 scores (ddof=1) with eps clamp
    const float mean = 0.25f * (score[0] + score[1] + score[2] + score[3]);
    const float c0 = score[0] - mean, c1 = score[1] - mean;
    const float c2 = score[2] - mean, c3 = score[3] - mean;
    const float var = (c0 * c0 + c1 * c1 + c2 * c2 + c3 * c3) * (1.0f / 3.0f);
    const float istd = frcp(fmaxf(fsqrt(var), 1e-4f));

    const float t0 = htanh(c0 * istd);
    const float t1 = htanh(c1 * istd);
    const float t2 = htanh(c2 * istd);
    const float t3 = htanh(c3 * istd);
    const float t4 = htanh(gq);
    const float t5 = htanh(gm);

    // 6-float channel pattern packed as 3 float4s (period 12 floats)
    const v4f w0 = {t0, t1, t2, t3};
    const v4f w1 = {t4, t5, t0, t1};
    const v4f w2 = {t2, t3, t4, t5};

    // comm: 48 floats = pattern repeated 4x (12 NT b128 stores)
    v4f* cp = (v4f*)(comm + (size_t)r * 48);
#pragma unroll
    for (int k = 0; k < 4; ++k) {
        __builtin_nontemporal_store(w0, cp + 3 * k + 0);
        __builtin_nontemporal_store(w1, cp + 3 * k + 1);
        __builtin_nontemporal_store(w2, cp + 3 * k + 2);
    }

    // state: first 32 floats of the same stream (8 NT b128 stores)
    v4f* sp = (v4f*)(state + (size_t)r * 32);
    __builtin_nontemporal_store(w0, sp + 0);
    __builtin_nontemporal_store(w1, sp + 1);
    __builtin_nontemporal_store(w2, sp + 2);
    __builtin_nontemporal_store(w0, sp + 3);
    __builtin_nontemporal_store(w1, sp + 4);
    __builtin_nontemporal_store(w2, sp + 5);
    __builtin_nontemporal_store(w0, sp + 6);
    __builtin_nontemporal_store(w1, sp + 7);
}

extern "C" void kernel_launch(void* const* d_in, const int* in_sizes, int n_in,
                              void* d_out, int out_size, void* d_ws, size_t ws_size,
                              hipStream_t stream)
{
    (void)n_in; (void)out_size; (void)d_ws; (void)ws_size;
    const float* in = (const float*)d_in[0];
    const int B = in_sizes[0] / ROW_F;          // 1,000,000

    float* comm  = (float*)d_out;               // B*48 floats
    float* state = (float*)d_out + (size_t)B * 48;  // B*32 floats

    const int grid = (B + BLK - 1) / BLK;
    hipLaunchKernelGGL(speaker_kernel, dim3(grid), dim3(BLK), 0, stream,
                       in, comm, state, B);
}